// DynamicConv_27522150432955
// MI455X (gfx1250) — compile-verified
//
#include <hip/hip_runtime.h>
#include <math.h>

typedef float v2f __attribute__((ext_vector_type(2)));
typedef float v8f __attribute__((ext_vector_type(8)));
typedef int   v4i __attribute__((ext_vector_type(4)));

#define CIP 12            // LDS x-tile inner (ci) pitch: 8 channels padded to 12
#define WKP 76            // LDS weight row pitch: 72 k-values padded to 76
#define XSZ (6 * 66 * CIP)  // 4752 floats per x buffer
#define WSZ (64 * WKP)      // 4864 floats per weight buffer

#define AS1 __attribute__((address_space(1)))
#define AS3 __attribute__((address_space(3)))

#if __has_builtin(__builtin_amdgcn_global_load_async_to_lds_b32) && \
    __has_builtin(__builtin_amdgcn_global_load_async_to_lds_b128)
#define HAVE_ASYNC_LDS 1
#else
#define HAVE_ASYNC_LDS 0
#endif

static __device__ __forceinline__ void wait_async0() {
#if __has_builtin(__builtin_amdgcn_s_wait_asynccnt)
  __builtin_amdgcn_s_wait_asynccnt(0);
#else
  asm volatile("s_wait_asynccnt 0" ::: "memory");
#endif
}

// ---------------------------------------------------------------------------
// Kernel 1: routing = sigmoid(mean_hw(x) @ w_route^T + b_route)   [B, E]
// ---------------------------------------------------------------------------
__global__ __launch_bounds__(256) void dynconv_routing_kernel(
    const float* __restrict__ x, const float* __restrict__ w_route,
    const float* __restrict__ b_route, float* __restrict__ routing)
{
  const int b = blockIdx.x;
  const int c = threadIdx.x;
  const float4* xv = (const float4*)(x + (((size_t)b * 256 + c) * 4096));
  float s = 0.0f;
#pragma unroll 4
  for (int i = 0; i < 1024; ++i) {
    float4 v = xv[i];
    s += (v.x + v.y) + (v.z + v.w);
  }
  __shared__ float pooled[256];
  pooled[c] = s * (1.0f / 4096.0f);
  __syncthreads();
  if (c < 4) {
    float acc = b_route[c];
    for (int i = 0; i < 256; ++i) acc += pooled[i] * w_route[c * 256 + i];
    routing[b * 4 + c] = 1.0f / (1.0f + __expf(-acc));
  }
}

// ---------------------------------------------------------------------------
// Kernel 2: per-sample combined weights, K-reordered (ci innermost):
//   wc2[b][co][r][ci] = sum_e routing[b,e] * w_experts[e][co][ci][r]
// ---------------------------------------------------------------------------
__global__ __launch_bounds__(256) void dynconv_combine_kernel(
    const float* __restrict__ w_experts, const float* __restrict__ routing,
    float* __restrict__ wc2)
{
  const int co = blockIdx.x;
  const int b  = blockIdx.y;
  const int ci = threadIdx.x;
  float acc[9] = {};
#pragma unroll
  for (int e = 0; e < 4; ++e) {
    const float rr = routing[b * 4 + e];
    const float* wp = w_experts + (((size_t)e * 256 + co) * 256 + ci) * 9;
#pragma unroll
    for (int k = 0; k < 9; ++k) acc[k] = fmaf(rr, wp[k], acc[k]);
  }
  float* op = wc2 + ((size_t)(b * 256 + co) * 9) * 256 + ci;
#pragma unroll
  for (int r = 0; r < 9; ++r) op[r * 256] = acc[r];
}

// ---------------------------------------------------------------------------
// Kernel 3: implicit-GEMM 3x3 conv (combined weights) + BN + SiLU.
// Grid (16 px-groups of 4 rows, 4 co-groups of 64, 16 batch); 256 thr = 8 waves.
// Wave (wm, wn) owns 16 co x 128 px = 8 f32 16x16 accumulator tiles.
// K = 2304 in chunks of 8 input channels; LDS double-buffered with
// GLOBAL_LOAD_ASYNC_TO_LDS (ASYNCcnt) when available.
// ---------------------------------------------------------------------------
__global__ __launch_bounds__(256) void dynconv_wmma_kernel(
    const float* __restrict__ x, const float* __restrict__ wc2,
    const float* __restrict__ bn_gamma, const float* __restrict__ bn_beta,
    const float* __restrict__ bn_mean, const float* __restrict__ bn_var,
    float* __restrict__ out)
{
#if HAVE_ASYNC_LDS
  __shared__ float sX[2 * XSZ];
  __shared__ float sW[2 * WSZ];
#else
  __shared__ float sX[XSZ];
  __shared__ float sW[WSZ];
#endif

  const int tid    = threadIdx.x;
  const int lane   = tid & 31;
  const int wave   = tid >> 5;
  const int wm     = wave & 3;   // co tile within block
  const int wn     = wave >> 2;  // pixel half (2 rows)
  const int lane15 = lane & 15;
  const int hi     = lane >> 4;
  const int hi2    = hi << 1;

  const int h0  = blockIdx.x * 4;
  const int co0 = blockIdx.y * 64;
  const int b   = blockIdx.z;

  // Zero all x buffers once: halo columns (w=-1,64) and out-of-image rows are
  // never written by staging, so they stay zero for every chunk.
#if HAVE_ASYNC_LDS
  for (int i = tid; i < 2 * XSZ; i += 256) sX[i] = 0.0f;
#else
  for (int i = tid; i < XSZ; i += 256) sX[i] = 0.0f;
#endif
  __syncthreads();

  const v8f vzero = {0, 0, 0, 0, 0, 0, 0, 0};
  v8f acc[8];
#pragma unroll
  for (int nt = 0; nt < 8; ++nt) acc[nt] = vzero;

  // -------- GEMM inner body over one staged chunk --------
  auto compute = [&](const float* bx, const float* bw) {
#pragma unroll
    for (int r = 0; r < 9; ++r) {
      const int kh = r / 3;
      const int kw = r - kh * 3;
#pragma unroll
      for (int c4 = 0; c4 < 2; ++c4) {
        const v2f afrag =
            *(const v2f*)&bw[(wm * 16 + lane15) * WKP + r * 8 + c4 * 4 + hi2];
#pragma unroll
        for (int nt = 0; nt < 8; ++nt) {
          const int rl = wn * 2 + (nt >> 2);
          const int cb = (nt & 3) * 16;
          const v2f bfrag = *(const v2f*)&bx[((rl + kh) * 66 + cb + lane15 + kw) * CIP +
                                             c4 * 4 + hi2];
          acc[nt] = __builtin_amdgcn_wmma_f32_16x16x4_f32(
              false, afrag, false, bfrag, (short)0, acc[nt], false, false);
        }
      }
    }
  };

#if HAVE_ASYNC_LDS
  // -------- async staging of one chunk into a given buffer --------
  auto issue_stage = [&](int chunk, float* bx, float* bw) {
    const int ci0 = chunk * 8;
    // x tile: 8 ci x 6 rows x 64 cols, transposed to ci-innermost (b32 scatter)
#pragma unroll
    for (int i = 0; i < 12; ++i) {
      const int ee   = tid + i * 256;    // 3072 elements
      const int col  = ee & 63;          // fastest -> coalesced global
      const int ci   = (ee >> 6) & 7;
      const int row  = ee >> 9;          // 0..5 -> image row h0-1+row
      const int rimg = h0 - 1 + row;
      if (rimg >= 0 && rimg < 64) {
        __builtin_amdgcn_global_load_async_to_lds_b32(
            (AS1 int*)(x + (((size_t)b * 256 + ci0 + ci) * 64 + rimg) * 64 + col),
            (AS3 int*)&bx[(row * 66 + 1 + col) * CIP + ci], 0, 0);
      }
    }
    // weight tile: 64 co x 9 r x 8 ci, contiguous 16B on both sides (b128)
#pragma unroll
    for (int i = 0; i < 5; ++i) {
      const int e = tid + i * 256;
      if (e < 1152) {
        const int c4 = e & 1;
        const int r  = (e >> 1) % 9;
        const int co = e / 18;
        __builtin_amdgcn_global_load_async_to_lds_b128(
            (AS1 v4i*)(wc2 + ((size_t)((b * 256 + co0 + co) * 9 + r)) * 256 + ci0 + c4 * 4),
            (AS3 v4i*)&bw[co * WKP + r * 8 + c4 * 4], 0, 0);
      }
    }
  };

  // Prologue: stage chunk 0 into buffer 0.
  issue_stage(0, &sX[0], &sW[0]);
  wait_async0();
  __syncthreads();

  int pb = 0;
  for (int chunk = 0; chunk < 32; ++chunk) {
    float* bx = &sX[pb * XSZ];
    float* bw = &sW[pb * WSZ];
    if (chunk + 1 < 32) issue_stage(chunk + 1, &sX[(pb ^ 1) * XSZ], &sW[(pb ^ 1) * WSZ]);
    compute(bx, bw);
    wait_async0();     // my chunk+1 async loads done
    __syncthreads();   // everyone done computing chunk & staging chunk+1
    pb ^= 1;
  }
#else
  // -------- synchronous fallback: register-staged, two barriers per chunk ----
  for (int chunk = 0; chunk < 32; ++chunk) {
    const int ci0 = chunk * 8;
#pragma unroll
    for (int i = 0; i < 3; ++i) {
      const int e    = tid + i * 256;
      const int grp  = e / 384;
      const int pos  = e % 384;
      const int col  = pos & 63;
      const int row  = pos >> 6;
      const int rimg = h0 - 1 + row;
      float4 v = make_float4(0.f, 0.f, 0.f, 0.f);
      if (rimg >= 0 && rimg < 64) {
        const float* xp = x + (((size_t)b * 256 + ci0 + grp * 4) * 64 + rimg) * 64 + col;
        v.x = xp[0];
        v.y = xp[4096];
        v.z = xp[8192];
        v.w = xp[12288];
      }
      *(float4*)&sX[(row * 66 + 1 + col) * CIP + grp * 4] = v;
    }
#pragma unroll
    for (int i = 0; i < 5; ++i) {
      const int e = tid + i * 256;
      if (e < 1152) {
        const int c4 = e & 1;
        const int r  = (e >> 1) % 9;
        const int co = e / 18;
        const float4 wv = *(const float4*)(
            wc2 + ((size_t)((b * 256 + co0 + co) * 9 + r)) * 256 + ci0 + c4 * 4);
        *(float4*)&sW[co * WKP + r * 8 + c4 * 4] = wv;
      }
    }
    __syncthreads();
    compute(sX, sW);
    __syncthreads();
  }
#endif

  // -------- fused BatchNorm (inference) + SiLU epilogue --------
  float inv[8], bet[8];
#pragma unroll
  for (int j = 0; j < 8; ++j) {
    const int co = co0 + wm * 16 + j + hi * 8;
    const float iv = bn_gamma[co] * rsqrtf(bn_var[co] + 1e-5f);
    inv[j] = iv;
    bet[j] = bn_beta[co] - bn_mean[co] * iv;
  }
#pragma unroll
  for (int nt = 0; nt < 8; ++nt) {
    const int rl = wn * 2 + (nt >> 2);
    const int h  = h0 + rl;
    const int w  = (nt & 3) * 16 + lane15;
#pragma unroll
    for (int j = 0; j < 8; ++j) {
      const int co = co0 + wm * 16 + j + hi * 8;  // C/D layout: VGPR j -> M = j + 8*hi
      const float v = acc[nt][j] * inv[j] + bet[j];
      const float o = v / (1.0f + __expf(-v));    // SiLU
      out[(((size_t)b * 256 + co) * 64 + h) * 64 + w] = o;
    }
  }
}

// ---------------------------------------------------------------------------
extern "C" void kernel_launch(void* const* d_in, const int* in_sizes, int n_in,
                              void* d_out, int out_size, void* d_ws, size_t ws_size,
                              hipStream_t stream) {
  (void)in_sizes; (void)n_in; (void)out_size; (void)ws_size;
  const float* x         = (const float*)d_in[0];
  const float* w_route   = (const float*)d_in[1];
  const float* b_route   = (const float*)d_in[2];
  const float* w_experts = (const float*)d_in[3];
  const float* bn_gamma  = (const float*)d_in[4];
  const float* bn_beta   = (const float*)d_in[5];
  const float* bn_mean   = (const float*)d_in[6];
  const float* bn_var    = (const float*)d_in[7];
  float* out = (float*)d_out;

  // workspace: [0..255] routing (64 used), then 16*256*9*256 combined weights (~37.75 MB)
  float* routing = (float*)d_ws;
  float* wc2     = (float*)d_ws + 256;

  dynconv_routing_kernel<<<16, 256, 0, stream>>>(x, w_route, b_route, routing);
  dynconv_combine_kernel<<<dim3(256, 16), 256, 0, stream>>>(w_experts, routing, wc2);
  dynconv_wmma_kernel<<<dim3(16, 4, 16), 256, 0, stream>>>(
      x, wc2, bn_gamma, bn_beta, bn_mean, bn_var, out);
}